// _QConv2d_MEDIUM_26817775796834
// MI455X (gfx1250) — compile-verified
//
#include <hip/hip_runtime.h>
#include <hip/hip_bf16.h>
#include <math.h>

// ---------------------------------------------------------------------------
// QConv2d on gfx1250:
//   kernel 1: fold each channel's (2 layers of Rot gates + CNOT ring) into a
//             fixed 16x16 complex unitary U_ic  (weights are data-independent)
//   kernel 2: per wave of 16 patches: Mottonen RY cascade (sqrt-only, no trig)
//             then S' = U_ic * S as complex 16x16x16 GEMM via
//             V_WMMA_F32_16X16X4_F32 (4 chained K=4 WMMAs per real matmul).
//   Cross-half-wave swaps done with v_permlanex16_b32 (pure VALU, no LDS pipe)
//   instead of ds_bpermute.
// ---------------------------------------------------------------------------

typedef float v2f __attribute__((ext_vector_type(2)));
typedef float v8f __attribute__((ext_vector_type(8)));

static __device__ __forceinline__ v8f wmma4(v2f a, v2f b, v8f c) {
  // D(16x16) += A(16x4) * B(4x16), f32
  return __builtin_amdgcn_wmma_f32_16x16x4_f32(false, a, false, b, (short)0, c,
                                               false, false);
}

// swap the two 16-lane halves of the wave: lane i <-> lane i^16 (VALU only)
static __device__ __forceinline__ float swap16(float v) {
  int i = __float_as_int(v);
  int r = __builtin_amdgcn_permlanex16(i, i, 0x76543210, 0xFEDCBA98,
                                       false, false);
  return __int_as_float(r);
}

static __device__ __forceinline__ void cs_from(float num, float den, float& c,
                                               float& s) {
  // cos/sin of (alpha/2) where alpha = 2*arcsin(sqrt(clip(num/den,0,1)))
  float inv = (den > 1e-20f) ? (1.0f / den) : 0.0f;
  float r2 = fminf(num * inv, 1.0f);
  s = sqrtf(r2);
  c = sqrtf(fmaxf(1.0f - r2, 0.0f));
}

// ------------------------- kernel 1: build unitaries ------------------------
// thread = ic*16 + col; each thread evolves one column of U_ic from e_col.
__global__ void _qconv_build_u(const float* __restrict__ w,
                               float* __restrict__ ws) {
  int tid = threadIdx.x;
  if (tid >= 64) return;
  int ic = tid >> 4;
  int col = tid & 15;
  float ur[16], ui[16];
#pragma unroll
  for (int i = 0; i < 16; ++i) {
    ur[i] = (i == col) ? 1.0f : 0.0f;
    ui[i] = 0.0f;
  }
#pragma unroll
  for (int l = 0; l < 2; ++l) {
#pragma unroll
    for (int wq = 0; wq < 4; ++wq) {
      int base = ((ic * 2 + l) * 4 + wq) * 3;
      float phi = w[base + 0];
      float th  = w[base + 1];
      float om  = w[base + 2];
      float ct = cosf(0.5f * th), st = sinf(0.5f * th);
      float ap = 0.5f * (phi + om), am = 0.5f * (phi - om);
      float cp = cosf(ap), sp = sinf(ap);
      float cm = cosf(am), sm = sinf(am);
      // Rot matrix: m00=e^{-i(phi+om)/2}ct, m01=-e^{i(phi-om)/2}st,
      //             m10=e^{-i(phi-om)/2}st, m11=e^{i(phi+om)/2}ct
      float m00r = cp * ct, m00i = -sp * ct;
      float m01r = -cm * st, m01i = -sm * st;
      float m10r = cm * st, m10i = -sm * st;
      float m11r = cp * ct, m11i = sp * ct;
      int mask = 1 << (3 - wq);  // wire wq <-> bit (3-wq)
#pragma unroll
      for (int i = 0; i < 16; ++i) {
        if (i & mask) continue;
        int j = i | mask;
        float u0r = ur[i], u0i = ui[i], u1r = ur[j], u1i = ui[j];
        ur[i] = m00r * u0r - m00i * u0i + m01r * u1r - m01i * u1i;
        ui[i] = m00r * u0i + m00i * u0r + m01r * u1i + m01i * u1r;
        ur[j] = m10r * u0r - m10i * u0i + m11r * u1r - m11i * u1i;
        ui[j] = m10r * u0i + m10i * u0r + m11r * u1i + m11i * u1r;
      }
    }
    int r = (l % 3) + 1;  // RANGES = [1, 2]
#pragma unroll
    for (int i = 0; i < 4; ++i) {
      int cmask = 1 << (3 - i);
      int tmask = 1 << (3 - ((i + r) & 3));
      float tr[16], ti[16];
#pragma unroll
      for (int idx = 0; idx < 16; ++idx) {
        int src = (idx & cmask) ? (idx ^ tmask) : idx;
        tr[idx] = ur[src];
        ti[idx] = ui[src];
      }
#pragma unroll
      for (int idx = 0; idx < 16; ++idx) {
        ur[idx] = tr[idx];
        ui[idx] = ti[idx];
      }
    }
  }
#pragma unroll
  for (int row = 0; row < 16; ++row) {
    ws[ic * 256 + row * 16 + col] = ur[row];          // Ur[ic][row][col]
    ws[1024 + ic * 256 + row * 16 + col] = ui[row];   // Ui[ic][row][col]
  }
}

// ----------------------------- kernel 2: main -------------------------------
__global__ __launch_bounds__(128) void _qconv_main(const float* __restrict__ x,
                                                   const float* __restrict__ U,
                                                   float* __restrict__ out) {
  const int lane = threadIdx.x & 31;
  const bool lo = lane < 16;  // lanes 0-15: j=0..7 ; lanes 16-31: j=8..15
  const int wave = (blockIdx.x * blockDim.x + threadIdx.x) >> 5;
  const int p = (wave << 4) + (lane & 15);  // patch index (b*64+y)*64+x
  const int xi = p & 63;
  const int yi = (p >> 6) & 63;
  const int bi = p >> 12;

  // state S[j][patch]: lane holds patch=lane&15, j = g + (lo?0:8)
  float sre[8], sim[8];
#pragma unroll
  for (int g = 0; g < 8; ++g) {
    sre[g] = 0.0f;
    sim[g] = 0.0f;
  }
  sre[0] = lo ? 1.0f : 0.0f;  // |0>

#pragma unroll
  for (int ic = 0; ic < 4; ++ic) {
    // ---- 3x3 patch squared values (pad = 0.01); normalization cancels ----
    float sq[9];
    const float* xc = x + (((size_t)bi * 4 + ic) << 12);
#pragma unroll
    for (int ki = 0; ki < 3; ++ki) {
#pragma unroll
      for (int kj = 0; kj < 3; ++kj) {
        int yy = yi - 1 + ki;
        int xx = xi - 1 + kj;
        bool inb = (yy >= 0) & (yy < 64) & (xx >= 0) & (xx < 64);
        float v = inb ? xc[(yy << 6) + xx] : 0.01f;
        sq[ki * 3 + kj] = v * v;
      }
    }
    // ---- binary tree of block sums (indices >= 9 are zero) ----
    float d3_0 = sq[0] + sq[1], n3_0 = sq[1];
    float d3_1 = sq[2] + sq[3], n3_1 = sq[3];
    float d3_2 = sq[4] + sq[5], n3_2 = sq[5];
    float d3_3 = sq[6] + sq[7], n3_3 = sq[7];
    float d3_4 = sq[8], n3_4 = 0.0f;
    float d2_0 = d3_0 + d3_1, n2_0 = d3_1;
    float d2_1 = d3_2 + d3_3, n2_1 = d3_3;
    float d2_2 = d3_4, n2_2 = 0.0f;
    float d1_0 = d2_0 + d2_1, n1_0 = d2_1;
    float d1_1 = d2_2, n1_1 = n2_2;
    float d0 = d1_0 + d1_1, n0v = d1_1;

    // per-lane-half selected rotation coefficients
    float c0, s0, c1, s1, c2a, s2a, c2b, s2b, c30, s30, c31, s31, c32, s32,
        c33, s33;
    cs_from(n0v, d0, c0, s0);
    cs_from(lo ? n1_0 : n1_1, lo ? d1_0 : d1_1, c1, s1);
    cs_from(lo ? n2_0 : n2_2, lo ? d2_0 : d2_2, c2a, s2a);
    cs_from(lo ? n2_1 : 0.0f, lo ? d2_1 : 0.0f, c2b, s2b);
    cs_from(lo ? n3_0 : n3_4, lo ? d3_0 : d3_4, c30, s30);
    cs_from(lo ? n3_1 : 0.0f, lo ? d3_1 : 0.0f, c31, s31);
    cs_from(lo ? n3_2 : 0.0f, lo ? d3_2 : 0.0f, c32, s32);
    cs_from(lo ? n3_3 : 0.0f, lo ? d3_3 : 0.0f, c33, s33);

    // ---- Mottonen cascade ----
    // t=0: mix j <-> j+8 (cross half-wave, v_permlanex16)
#pragma unroll
    for (int g = 0; g < 8; ++g) {
      float pr = swap16(sre[g]);
      float pi = swap16(sim[g]);
      sre[g] = lo ? (c0 * sre[g] - s0 * pr) : (s0 * pr + c0 * sre[g]);
      sim[g] = lo ? (c0 * sim[g] - s0 * pi) : (s0 * pi + c0 * sim[g]);
    }
    // t=1: pairs (g, g+4)
#pragma unroll
    for (int g = 0; g < 4; ++g) {
      float r0 = sre[g], r1 = sre[g + 4], i0 = sim[g], i1 = sim[g + 4];
      sre[g] = c1 * r0 - s1 * r1;
      sre[g + 4] = s1 * r0 + c1 * r1;
      sim[g] = c1 * i0 - s1 * i1;
      sim[g + 4] = s1 * i0 + c1 * i1;
    }
    // t=2: pairs (0,2),(1,3) [c2a] ; (4,6),(5,7) [c2b]
#pragma unroll
    for (int g = 0; g < 2; ++g) {
      float r0 = sre[g], r1 = sre[g + 2], i0 = sim[g], i1 = sim[g + 2];
      sre[g] = c2a * r0 - s2a * r1;
      sre[g + 2] = s2a * r0 + c2a * r1;
      sim[g] = c2a * i0 - s2a * i1;
      sim[g + 2] = s2a * i0 + c2a * i1;
      float r4 = sre[g + 4], r6 = sre[g + 6], i4 = sim[g + 4], i6 = sim[g + 6];
      sre[g + 4] = c2b * r4 - s2b * r6;
      sre[g + 6] = s2b * r4 + c2b * r6;
      sim[g + 4] = c2b * i4 - s2b * i6;
      sim[g + 6] = s2b * i4 + c2b * i6;
    }
    // t=3: pairs (0,1) (2,3) (4,5) (6,7)
    {
      float cc[4] = {c30, c31, c32, c33};
      float ss[4] = {s30, s31, s32, s33};
#pragma unroll
      for (int m2 = 0; m2 < 4; ++m2) {
        float r0 = sre[2 * m2], r1 = sre[2 * m2 + 1];
        float i0 = sim[2 * m2], i1 = sim[2 * m2 + 1];
        sre[2 * m2] = cc[m2] * r0 - ss[m2] * r1;
        sre[2 * m2 + 1] = ss[m2] * r0 + cc[m2] * r1;
        sim[2 * m2] = cc[m2] * i0 - ss[m2] * i1;
        sim[2 * m2 + 1] = ss[m2] * i0 + cc[m2] * i1;
      }
    }

    // ---- S' = U_ic * S, complex, via WMMA f32 16x16x4 ----
    const int m = lane & 15;
    const int kadd = lo ? 0 : 2;  // A/B operand K striping across half-waves
    v2f Ar[4], Ai[4];
    const float* Urb = U + ic * 256 + m * 16;
    const float* Uib = U + 1024 + ic * 256 + m * 16;
#pragma unroll
    for (int q = 0; q < 4; ++q) {
      int kb = q * 4 + kadd;
      Ar[q] = *(const v2f*)(Urb + kb);
      Ai[q] = *(const v2f*)(Uib + kb);
    }
    v2f Br[4], Bi[4];
    {
      float xr0 = swap16(sre[0]), xr1 = swap16(sre[1]);
      float xr2 = swap16(sre[2]), xr3 = swap16(sre[3]);
      float xr4 = swap16(sre[4]), xr5 = swap16(sre[5]);
      float xr6 = swap16(sre[6]), xr7 = swap16(sre[7]);
      Br[0].x = lo ? sre[0] : xr2;  Br[0].y = lo ? sre[1] : xr3;   // K 0..3
      Br[1].x = lo ? sre[4] : xr6;  Br[1].y = lo ? sre[5] : xr7;   // K 4..7
      Br[2].x = lo ? xr0 : sre[2];  Br[2].y = lo ? xr1 : sre[3];   // K 8..11
      Br[3].x = lo ? xr4 : sre[6];  Br[3].y = lo ? xr5 : sre[7];   // K 12..15
      float xi0 = swap16(sim[0]), xi1 = swap16(sim[1]);
      float xi2 = swap16(sim[2]), xi3 = swap16(sim[3]);
      float xi4 = swap16(sim[4]), xi5 = swap16(sim[5]);
      float xi6 = swap16(sim[6]), xi7 = swap16(sim[7]);
      Bi[0].x = lo ? sim[0] : xi2;  Bi[0].y = lo ? sim[1] : xi3;
      Bi[1].x = lo ? sim[4] : xi6;  Bi[1].y = lo ? sim[5] : xi7;
      Bi[2].x = lo ? xi0 : sim[2];  Bi[2].y = lo ? xi1 : sim[3];
      Bi[3].x = lo ? xi4 : sim[6];  Bi[3].y = lo ? xi5 : sim[7];
    }
    v8f accR = {0.f, 0.f, 0.f, 0.f, 0.f, 0.f, 0.f, 0.f};
    v8f accM = {0.f, 0.f, 0.f, 0.f, 0.f, 0.f, 0.f, 0.f};
    v8f accI = {0.f, 0.f, 0.f, 0.f, 0.f, 0.f, 0.f, 0.f};
#pragma unroll
    for (int q = 0; q < 4; ++q) accR = wmma4(Ar[q], Br[q], accR);  // Ur*Sr
#pragma unroll
    for (int q = 0; q < 4; ++q) accM = wmma4(Ai[q], Bi[q], accM);  // Ui*Si
#pragma unroll
    for (int q = 0; q < 4; ++q) accI = wmma4(Ar[q], Bi[q], accI);  // Ur*Si
#pragma unroll
    for (int q = 0; q < 4; ++q) accI = wmma4(Ai[q], Br[q], accI);  // +Ui*Sr
#pragma unroll
    for (int g = 0; g < 8; ++g) {
      sre[g] = accR[g] - accM[g];
      sim[g] = accI[g];
    }
  }

  // ---- probs -> clip -> output (b, 16, 64, 64) ----
  const int obase = (bi << 16) + (yi << 6) + xi;  // bi*16*4096 + y*64 + x
#pragma unroll
  for (int g = 0; g < 8; ++g) {
    int j = g + (lo ? 0 : 8);
    float pv = (sre[g] * sre[g] + sim[g] * sim[g]) * 8.0f;
    out[obase + (j << 12)] = fminf(pv, 1.0f);
  }
}

// ---------------------------------------------------------------------------
extern "C" void kernel_launch(void* const* d_in, const int* in_sizes, int n_in,
                              void* d_out, int out_size, void* d_ws,
                              size_t ws_size, hipStream_t stream) {
  const float* x = (const float*)d_in[0];        // (32,4,64,64)
  const float* w = (const float*)d_in[1];        // (4,2,4,3)
  float* out = (float*)d_out;                    // (32,16,64,64)
  float* U = (float*)d_ws;                       // 2048 floats: Ur[4][16][16], Ui[4][16][16]

  _qconv_build_u<<<1, 64, 0, stream>>>(w, U);

  const int n_patches = 32 * 64 * 64;            // 131072
  const int waves = n_patches / 16;              // 8192 waves, 16 patches each
  _qconv_main<<<waves / 4, 128, 0, stream>>>(x, U, out);
}